// SmoothOhemLoss_60481729462505
// MI455X (gfx1250) — compile-verified
//
#include <hip/hip_runtime.h>
#include <hip/hip_bf16.h>

// ---------------------------------------------------------------------------
// SmoothL1 + OHEM loss for MI455X (gfx1250).
// Memory-bound: 105MB reads ~= 4.5us at 23.3 TB/s. Top-k via 3-level radix
// histogram select on float bits (monotone for non-negative floats); the
// refinement passes re-read inputs from the 192MB L2 (fully resident).
// Wave-level stat reductions use V_WMMA_F32_16X16X4_F32 with a ones matrix.
// ---------------------------------------------------------------------------

typedef __attribute__((ext_vector_type(2))) float v2f;
typedef __attribute__((ext_vector_type(8))) float v8f;

#define TPB 256
#define EPB 4096      // elements per block in streaming kernels
#define NB0 2048      // level-0 bins: bits[30:20]  (sign always 0 for loss>=0)
#define NB12 1024     // level-1/2 bins: bits[19:10], bits[9:0]

struct SelState {
  unsigned prefix;   // accumulated cut bits (c0, then (c0<<10)|c1)
  int kRem;          // remaining k inside the current cut bin
  unsigned Cgt;      // count of values strictly above cut so far
  float Sgt;         // sum of values strictly above cut so far
  int k;             // total k for this sample
  int done;          // 1 => total negatives < k (take everything)
  float posTerm;     // pos_mean if num_pos>0 else 0
  int pad;
};

__device__ __forceinline__ float smooth_l1_w(float p, float g, float w) {
  float d = fabsf(p - g);
  float l = (d < 1.0f) ? (0.5f * d * d) : (d - 0.5f);
  return l * w;   // w > 0 => loss >= 0 => sign bit 0 => uint bits monotone
}

// Reduce two per-lane sub-accumulators across the wave (32 lanes).
// One V_WMMA_F32_16X16X4_F32 with B=ones collapses A's 64 values into 16
// identical-per-column row sums (D[m][n] = a0[m]+a1[m]+a0[m+16]+a1[m+16]).
// C-layout: lanes 0-15 hold rows 0..7 in d[0..7], lanes 16-31 rows 8..15,
// so 7 adds + one shfl_xor(16) finish the full wave sum on every lane.
__device__ __forceinline__ float wave_sum_pair(float a0, float a1) {
#if __has_builtin(__builtin_amdgcn_wmma_f32_16x16x4_f32)
  v2f a; a.x = a0; a.y = a1;
  v2f b; b.x = 1.0f; b.y = 1.0f;
  v8f c = {};
  v8f d = __builtin_amdgcn_wmma_f32_16x16x4_f32(false, a, false, b,
                                                (short)0, c, false, false);
  float r = d[0] + d[1] + d[2] + d[3] + d[4] + d[5] + d[6] + d[7];
  r += __shfl_xor(r, 16, 32);
  return r;
#else
  float r = a0 + a1;
  for (int o = 16; o > 0; o >>= 1) r += __shfl_xor(r, o, 32);
  return r;
#endif
}

// ---------------------------------------------------------------------------
// Kernel 1: streaming pass. Per-sample stats + level-0 histogram.
// grid = (ceil(N/4096), B), 256 threads.
// ---------------------------------------------------------------------------
__global__ void __launch_bounds__(TPB)
ohem_main_pass(const float* __restrict__ p, const float* __restrict__ g,
               const float* __restrict__ ig, const float* __restrict__ w,
               float* __restrict__ stats, unsigned* __restrict__ h0c,
               float* __restrict__ h0s, int N)
{
  __shared__ unsigned lc[NB0];
  __shared__ float ls[NB0];
  const int t = threadIdx.x;
  const int s = blockIdx.y;
  for (int i = t; i < NB0; i += TPB) { lc[i] = 0u; ls[i] = 0.0f; }
  __syncthreads();

  const size_t base = (size_t)s * (size_t)N;
  const float4* p4 = (const float4*)(p + base);
  const float4* g4 = (const float4*)(g + base);
  const float4* i4 = (const float4*)(ig + base);
  const float4* w4 = (const float4*)(w + base);

  float ps[2] = {0.f, 0.f}, np[2] = {0.f, 0.f}, na[2] = {0.f, 0.f};

  const int q0 = blockIdx.x * (EPB / 4);
#pragma unroll
  for (int j = 0; j < 4; ++j) {
    const int q = q0 + j * TPB + t;
    if (q * 4 < N) {
      float4 pv = p4[q], gv = g4[q], iv = i4[q], wv = w4[q];
      float pa[4] = {pv.x, pv.y, pv.z, pv.w};
      float ga[4] = {gv.x, gv.y, gv.z, gv.w};
      float ia[4] = {iv.x, iv.y, iv.z, iv.w};
      float wa[4] = {wv.x, wv.y, wv.z, wv.w};
#pragma unroll
      for (int e = 0; e < 4; ++e) {
        float G = ga[e], IG = ia[e];
        float L = smooth_l1_w(pa[e], G, wa[e]);
        na[e & 1] += IG;                       // ig in {0,1}; exact in f32
        if (G > 0.0f) { ps[e & 1] += L; np[e & 1] += 1.0f; }
        if ((G == 0.0f) && (IG > 0.0f)) {
          unsigned bits = __float_as_uint(L);
          unsigned bin = bits >> 20;           // 11 bits, sign==0
          atomicAdd(&lc[bin], 1u);
          atomicAdd(&ls[bin], L);
        }
      }
    }
  }
  __syncthreads();

  // Merge only populated bins to cut global atomic traffic.
  for (int i = t; i < NB0; i += TPB) {
    unsigned c = lc[i];
    if (c) {
      atomicAdd(&h0c[s * NB0 + i], c);
      atomicAdd(&h0s[s * NB0 + i], ls[i]);
    }
  }

  // WMMA-based wave reductions (EXEC is all-ones here: no divergence).
  float psW = wave_sum_pair(ps[0], ps[1]);
  float npW = wave_sum_pair(np[0], np[1]);
  float naW = wave_sum_pair(na[0], na[1]);
  if ((t & 31) == 0) {
    atomicAdd(&stats[s * 4 + 0], psW);
    atomicAdd(&stats[s * 4 + 1], npW);
    atomicAdd(&stats[s * 4 + 2], naW);
  }
}

// ---------------------------------------------------------------------------
// Refinement histogram pass (levels 1 & 2). Re-reads inputs (L2-resident),
// histograms only negatives whose high bits match the current cut prefix.
// predShift = 20 (level 1) or 10 (level 2); bin = bits >> (predShift-10).
// ---------------------------------------------------------------------------
__global__ void __launch_bounds__(TPB)
ohem_hist_pass(const float* __restrict__ p, const float* __restrict__ g,
               const float* __restrict__ ig, const float* __restrict__ w,
               const SelState* __restrict__ st, unsigned* __restrict__ gc,
               float* __restrict__ gs, int N, int predShift)
{
  const int s = blockIdx.y;
  SelState S = st[s];
  if (S.done) return;
  const unsigned prefix = S.prefix;
  const int binShift = predShift - 10;

  __shared__ unsigned lc[NB12];
  __shared__ float ls[NB12];
  const int t = threadIdx.x;
  for (int i = t; i < NB12; i += TPB) { lc[i] = 0u; ls[i] = 0.0f; }
  __syncthreads();

  const size_t base = (size_t)s * (size_t)N;
  const float4* p4 = (const float4*)(p + base);
  const float4* g4 = (const float4*)(g + base);
  const float4* i4 = (const float4*)(ig + base);
  const float4* w4 = (const float4*)(w + base);

  const int q0 = blockIdx.x * (EPB / 4);
#pragma unroll
  for (int j = 0; j < 4; ++j) {
    const int q = q0 + j * TPB + t;
    if (q * 4 < N) {
      float4 pv = p4[q], gv = g4[q], iv = i4[q], wv = w4[q];
      float pa[4] = {pv.x, pv.y, pv.z, pv.w};
      float ga[4] = {gv.x, gv.y, gv.z, gv.w};
      float ia[4] = {iv.x, iv.y, iv.z, iv.w};
      float wa[4] = {wv.x, wv.y, wv.z, wv.w};
#pragma unroll
      for (int e = 0; e < 4; ++e) {
        if ((ga[e] == 0.0f) && (ia[e] > 0.0f)) {
          float L = smooth_l1_w(pa[e], ga[e], wa[e]);
          unsigned bits = __float_as_uint(L);
          if ((bits >> predShift) == prefix) {
            unsigned bin = (bits >> binShift) & (NB12 - 1);
            atomicAdd(&lc[bin], 1u);
            atomicAdd(&ls[bin], L);
          }
        }
      }
    }
  }
  __syncthreads();
  for (int i = t; i < NB12; i += TPB) {
    unsigned c = lc[i];
    if (c) {
      atomicAdd(&gc[s * NB12 + i], c);
      atomicAdd(&gs[s * NB12 + i], ls[i]);
    }
  }
}

// ---------------------------------------------------------------------------
// Cut-bin selection. One block per sample. Parallel suffix scan over segment
// sums (Hillis-Steele) -> unique crossing segment -> <=8-step inner scan.
// Level 2 finalizes: topk = Sgt + (k - Cgt) * threshold (exact w.r.t. ties).
// ---------------------------------------------------------------------------
__global__ void __launch_bounds__(TPB)
ohem_select_pass(int level, int nbins, const float* __restrict__ stats,
                 const unsigned* __restrict__ gc, const float* __restrict__ gs,
                 SelState* __restrict__ st, float* __restrict__ out, float invB)
{
  const int s = blockIdx.x;
  const int t = threadIdx.x;
  __shared__ unsigned lc[NB0];
  __shared__ float ls[NB0];
  __shared__ unsigned segC[TPB];
  __shared__ float segS[TPB];
  __shared__ int winner;

  int kEff, kTot;
  unsigned Cbase = 0, prefix = 0;
  float Sbase = 0.f, posTerm = 0.f;

  if (level == 0) {
    float posSum = stats[s * 4 + 0];
    int numPos = (int)(stats[s * 4 + 1] + 0.5f);
    int numAll = (int)(stats[s * 4 + 2] + 0.5f);
    kTot = (numPos > 0) ? min(3 * numPos, numAll - numPos) : 500;
    kEff = kTot;
    posTerm = (numPos > 0) ? posSum / (float)max(numPos, 1) : 0.f;
  } else {
    SelState S = st[s];
    if (S.done) {
      if (level == 2 && t == 0) {
        float psamp = S.posTerm + S.Sgt / (float)S.k;  // took everything
        atomicAdd(out, psamp * invB);
      }
      return;
    }
    kEff = S.kRem; Cbase = S.Cgt; Sbase = S.Sgt;
    prefix = S.prefix; posTerm = S.posTerm; kTot = S.k;
  }

  for (int i = t; i < nbins; i += TPB) {
    lc[i] = gc[s * nbins + i];
    ls[i] = gs[s * nbins + i];
  }
  __syncthreads();

  const int segSize = nbins / TPB;   // 8 (L0) or 4 (L1/L2)
  unsigned myC = 0; float myS = 0.f;
  for (int i = 0; i < segSize; ++i) { myC += lc[t * segSize + i]; myS += ls[t * segSize + i]; }
  segC[t] = myC; segS[t] = myS;
  __syncthreads();
  for (int off = 1; off < TPB; off <<= 1) {       // inclusive suffix scan
    unsigned c = (t + off < TPB) ? segC[t + off] : 0u;
    float sv = (t + off < TPB) ? segS[t + off] : 0.f;
    __syncthreads();
    segC[t] += c; segS[t] += sv;
    __syncthreads();
  }
  const unsigned Cex = segC[t] - myC;   // counts in segments strictly above t
  const float Sex = segS[t] - myS;

  if (t == 0) winner = -1;
  __syncthreads();
  bool cond = ((long)Cex < (long)kEff) && ((long)Cex + (long)myC >= (long)kEff);
  if (cond) winner = t;                 // unique when 1 <= kEff <= total
  __syncthreads();
  const int win = winner;

  if (t == win) {
    unsigned cum = Cex; float ssum = Sex; int cut = 0;
    for (int i = segSize - 1; i >= 0; --i) {
      int idx = win * segSize + i;
      unsigned c = lc[idx];
      if ((long)cum + (long)c >= (long)kEff) { cut = idx; break; }
      cum += c; ssum += ls[idx];
    }
    if (level < 2) {
      SelState O;
      O.done = 0;
      O.prefix = (level == 0) ? (unsigned)cut : ((prefix << 10) | (unsigned)cut);
      O.kRem = kEff - (int)cum;
      O.Cgt = Cbase + cum;
      O.Sgt = Sbase + ssum;
      O.k = kTot; O.posTerm = posTerm; O.pad = 0;
      st[s] = O;
    } else {
      unsigned T = (prefix << 10) | (unsigned)cut;   // exact threshold bits
      float thr = __uint_as_float(T);
      float Sgt = Sbase + ssum;
      float rem = (float)(kEff - (int)cum);          // ties at threshold
      float topk = Sgt + rem * thr;
      float psamp = posTerm + topk / (float)kTot;
      atomicAdd(out, psamp * invB);
    }
  }
  if (t == 0 && win < 0) {              // total < kEff: take everything
    if (level < 2) {
      SelState O;
      O.done = 1; O.prefix = 0; O.kRem = 0;
      O.Cgt = Cbase + segC[0]; O.Sgt = Sbase + segS[0];
      O.k = kTot; O.posTerm = posTerm; O.pad = 0;
      st[s] = O;
    } else {
      float psamp = posTerm + (Sbase + segS[0]) / (float)kTot;
      atomicAdd(out, psamp * invB);
    }
  }
}

// ---------------------------------------------------------------------------
extern "C" void kernel_launch(void* const* d_in, const int* in_sizes, int n_in,
                              void* d_out, int out_size, void* d_ws, size_t ws_size,
                              hipStream_t stream) {
  const float* p  = (const float*)d_in[0];
  const float* g  = (const float*)d_in[1];
  const float* ig = (const float*)d_in[2];
  const float* w  = (const float*)d_in[3];
  float* out = (float*)d_out;

  const int Bn = 16;
  const int N = in_sizes[0] / Bn;

  // Workspace layout (~525 KB total).
  char* ws = (char*)d_ws;
  float*    stats = (float*)ws;                              // 16*4 f32
  SelState* st    = (SelState*)(ws + 256);                   // 16*32 B
  char* cur = ws + 1024;
  unsigned* h0c = (unsigned*)cur; cur += (size_t)Bn * NB0 * 4;
  float*    h0s = (float*)cur;    cur += (size_t)Bn * NB0 * 4;
  unsigned* h1c = (unsigned*)cur; cur += (size_t)Bn * NB12 * 4;
  float*    h1s = (float*)cur;    cur += (size_t)Bn * NB12 * 4;
  unsigned* h2c = (unsigned*)cur; cur += (size_t)Bn * NB12 * 4;
  float*    h2s = (float*)cur;    cur += (size_t)Bn * NB12 * 4;
  size_t used = (size_t)(cur - ws);

  hipMemsetAsync(d_ws, 0, used, stream);
  hipMemsetAsync(d_out, 0, (size_t)out_size * sizeof(float), stream);

  const int blocksX = (N + EPB - 1) / EPB;   // 100 for N=409600
  dim3 gridS(blocksX, Bn);
  const float invB = 1.0f / (float)Bn;

  ohem_main_pass<<<gridS, TPB, 0, stream>>>(p, g, ig, w, stats, h0c, h0s, N);
  ohem_select_pass<<<Bn, TPB, 0, stream>>>(0, NB0, stats, h0c, h0s, st, out, invB);
  ohem_hist_pass<<<gridS, TPB, 0, stream>>>(p, g, ig, w, st, h1c, h1s, N, 20);
  ohem_select_pass<<<Bn, TPB, 0, stream>>>(1, NB12, stats, h1c, h1s, st, out, invB);
  ohem_hist_pass<<<gridS, TPB, 0, stream>>>(p, g, ig, w, st, h2c, h2s, N, 10);
  ohem_select_pass<<<Bn, TPB, 0, stream>>>(2, NB12, stats, h2c, h2s, st, out, invB);
}